// MaxMinSampler_18322330485333
// MI455X (gfx1250) — compile-verified
//
#include <hip/hip_runtime.h>
#include <math.h>

// ---------------------------------------------------------------------------
// MaxMinSampler for MI455X (gfx1250, wave32)
//   Phase 1: feats = features @ W.T + b  via V_WMMA_F32_16X16X4_F32 (fp32 WMMA)
//   Phase 2: row L2-normalize in place
//   Phase 3: farthest-point sampling, 512 rounds, fn resident in 192MB L2
// ---------------------------------------------------------------------------

#define IN_DIM  768
#define RED_DIM 128
#define KSEL    512

typedef float v2f __attribute__((ext_vector_type(2)));
typedef float v8f __attribute__((ext_vector_type(8)));

// ---------------------------------------------------------------------------
// GEMM: block = 256 threads (8 waves). Block handles rows [16*bid, 16*bid+16),
// wave w handles output columns [16w, 16w+16). Accumulate over K=768 in
// chunks of 4 with fp32 WMMA (192 wmma ops per wave).
//
// A fragment layout (16x4 f32, ISA 7.12.2): lane&15 = M row; lanes 0-15 hold
// K = kk+0 (v0) / kk+1 (v1); lanes 16-31 hold K = kk+2 / kk+3.
// B fragment (4x16): lane&15 = N col, same K striping. B[k][n] = W[n][k].
// C/D layout: lane&15 = N; VGPR r holds M = r (lanes 0-15) or r+8 (16-31).
//
// N is launched as exactly N/16 blocks (100000 = 6250*16), so every row is
// valid: the epilogue stores are unconditional (no EXEC churn after WMMA).
// ---------------------------------------------------------------------------
__global__ __launch_bounds__(256) void gemm_wmma_f32(
    const float* __restrict__ A,     // [N, 768] features, row major
    const float* __restrict__ W,     // [128, 768] row major
    const float* __restrict__ bias,  // [128]
    float* __restrict__ out,         // [N, 128] feats, row major
    int N)
{
    const int row0 = blockIdx.x * 16;
    const int wave = threadIdx.x >> 5;   // 0..7
    const int lane = threadIdx.x & 31;
    const int col0 = wave * 16;

    const int m    = lane & 15;
    const int koff = (lane >> 4) * 2;    // 0 for lanes 0-15, 2 for 16-31

    const float* __restrict__ arow = A + (size_t)(row0 + m) * IN_DIM + koff;
    const float* __restrict__ wrow = W + (size_t)(col0 + m) * IN_DIM + koff;

    v8f acc = {};
#pragma unroll 8
    for (int kk = 0; kk < IN_DIM; kk += 4) {
        v2f a, b;
        a.x = arow[kk];
        a.y = arow[kk + 1];
        b.x = wrow[kk];
        b.y = wrow[kk + 1];
        // (neg_a, A, neg_b, B, c_mod, C, reuse_a, reuse_b)
        acc = __builtin_amdgcn_wmma_f32_16x16x4_f32(
            false, a, false, b, (short)0, acc, false, false);
    }

    const int n     = lane & 15;
    const int rbase = (lane >> 4) * 8;
    const float bv  = bias[col0 + n];
    float* __restrict__ obase =
        out + (size_t)(row0 + rbase) * RED_DIM + col0 + n;
#pragma unroll
    for (int r = 0; r < 8; ++r) {
        obase[(size_t)r * RED_DIM] = acc[r] + bv;
    }
}

// ---------------------------------------------------------------------------
// Row L2 normalize in place. One block (128 threads) per row.
// ---------------------------------------------------------------------------
__global__ __launch_bounds__(128) void normalize_rows(
    float* __restrict__ feats, int N)
{
    __shared__ float red[128];
    const int row = blockIdx.x;
    const int t   = threadIdx.x;
    const float v = feats[(size_t)row * RED_DIM + t];
    red[t] = v * v;
    __syncthreads();
#pragma unroll
    for (int s = 64; s > 0; s >>= 1) {
        if (t < s) red[t] += red[t + s];
        __syncthreads();
    }
    float nrm = sqrtf(red[0]);
    nrm = fmaxf(nrm, 1e-12f);
    feats[(size_t)row * RED_DIM + t] = v / nrm;
}

// ---------------------------------------------------------------------------
// Block-level argmax over attention scores (first-index tie break).
// ---------------------------------------------------------------------------
__global__ __launch_bounds__(256) void argmax_part(
    const float* __restrict__ x, int N,
    float* __restrict__ pval, int* __restrict__ pidx)
{
    __shared__ float sv[256];
    __shared__ int   si[256];
    const int t = threadIdx.x;
    const int j = blockIdx.x * 256 + t;
    sv[t] = (j < N) ? x[j] : -__builtin_inff();
    si[t] = j;
    __syncthreads();
#pragma unroll
    for (int s = 128; s > 0; s >>= 1) {
        if (t < s) {
            if (sv[t + s] > sv[t] ||
                (sv[t + s] == sv[t] && si[t + s] < si[t])) {
                sv[t] = sv[t + s];
                si[t] = si[t + s];
            }
        }
        __syncthreads();
    }
    if (t == 0) { pval[blockIdx.x] = sv[0]; pidx[blockIdx.x] = si[0]; }
}

// ---------------------------------------------------------------------------
// FPS distance sweep + block argmax. Selected row is staged in LDS.
//   init==1: min_d[j] = d(j, cur)   (with min_d[cur] = -inf)
//   init==0: min_d[j] = min(min_d[j], d(j, cur))
// fn (51.2 MB) stays resident in the 192 MB L2 across all 511 sweeps.
// ---------------------------------------------------------------------------
__global__ __launch_bounds__(256) void fps_part(
    const float* __restrict__ fn,   // [N, 128] normalized
    float* __restrict__ mind,       // [N]
    const int* __restrict__ cur,    // [1]
    int N, int init,
    float* __restrict__ pval, int* __restrict__ pidx)
{
    __shared__ float srow[RED_DIM];
    __shared__ float sv[256];
    __shared__ int   si[256];

    const int t = threadIdx.x;
    const int c = cur[0];
    if (t < RED_DIM) srow[t] = fn[(size_t)c * RED_DIM + t];
    __syncthreads();

    const int j = blockIdx.x * 256 + t;
    float key = -__builtin_inff();
    if (j < N) {
        const float4* __restrict__ rp = (const float4*)(fn + (size_t)j * RED_DIM);
        const float4* __restrict__ sp = (const float4*)srow;
        float acc = 0.0f;
#pragma unroll
        for (int q = 0; q < RED_DIM / 4; ++q) {
            const float4 a = rp[q];
            const float4 b = sp[q];
            const float dx = a.x - b.x, dy = a.y - b.y;
            const float dz = a.z - b.z, dw = a.w - b.w;
            acc = fmaf(dx, dx, acc);
            acc = fmaf(dy, dy, acc);
            acc = fmaf(dz, dz, acc);
            acc = fmaf(dw, dw, acc);
        }
        const float d = sqrtf(fmaxf(acc, 0.0f));
        float nm;
        if (init) nm = (j == c) ? -__builtin_inff() : d;
        else      nm = fminf(mind[j], d);
        mind[j] = nm;
        key = nm;
    }

    sv[t] = key;
    si[t] = j;
    __syncthreads();
#pragma unroll
    for (int s = 128; s > 0; s >>= 1) {
        if (t < s) {
            if (sv[t + s] > sv[t] ||
                (sv[t + s] == sv[t] && si[t + s] < si[t])) {
                sv[t] = sv[t + s];
                si[t] = si[t + s];
            }
        }
        __syncthreads();
    }
    if (t == 0) { pval[blockIdx.x] = sv[0]; pidx[blockIdx.x] = si[0]; }
}

// ---------------------------------------------------------------------------
// Final argmax over block partials. Records pick, updates cur, poisons min_d.
// nb <= 512 so one load per thread suffices.
// ---------------------------------------------------------------------------
__global__ __launch_bounds__(512) void fps_final(
    const float* __restrict__ pval, const int* __restrict__ pidx, int nb,
    int* __restrict__ cur, float* __restrict__ out, int slot,
    float* __restrict__ mind, int set_inf)
{
    __shared__ float sv[512];
    __shared__ int   si[512];
    const int t = threadIdx.x;
    if (t < nb) { sv[t] = pval[t]; si[t] = pidx[t]; }
    else        { sv[t] = -__builtin_inff(); si[t] = 0x7fffffff; }
    __syncthreads();
#pragma unroll
    for (int s = 256; s > 0; s >>= 1) {
        if (t < s) {
            if (sv[t + s] > sv[t] ||
                (sv[t + s] == sv[t] && si[t + s] < si[t])) {
                sv[t] = sv[t + s];
                si[t] = si[t + s];
            }
        }
        __syncthreads();
    }
    if (t == 0) {
        const int sel = si[0];
        cur[0]    = sel;
        out[slot] = (float)sel;   // indices < 2^24: exact in fp32
        if (set_inf) mind[sel] = -__builtin_inff();
    }
}

// ---------------------------------------------------------------------------
// Host launch: graph-capture safe (only kernel launches on `stream`).
// ---------------------------------------------------------------------------
extern "C" void kernel_launch(void* const* d_in, const int* in_sizes, int n_in,
                              void* d_out, int out_size, void* d_ws, size_t ws_size,
                              hipStream_t stream)
{
    const float* features = (const float*)d_in[0];   // [N, 768]
    const float* attn     = (const float*)d_in[1];   // [N]
    const float* W        = (const float*)d_in[2];   // [128, 768]
    const float* bias     = (const float*)d_in[3];   // [128]
    const int N = in_sizes[0] / IN_DIM;              // 100000

    float* out = (float*)d_out;                      // 512 selected indices

    // workspace layout
    float* feats = (float*)d_ws;                     // N*128 floats
    float* mind  = feats + (size_t)N * RED_DIM;      // N floats
    const int BA = (N + 255) / 256;                  // 391 blocks
    float* pval  = mind + N;                         // BA floats
    int*   pidx  = (int*)(pval + BA);                // BA ints
    int*   cur   = pidx + BA;                        // 1 int

    // Phase 1: fp32 WMMA GEMM (N is a multiple of 16: 100000 = 6250*16)
    gemm_wmma_f32<<<N / 16, 256, 0, stream>>>(features, W, bias, feats, N);

    // Phase 2: row L2 normalize
    normalize_rows<<<N, 128, 0, stream>>>(feats, N);

    // Phase 3: FPS. Round 0: argmax of attention scores.
    argmax_part<<<BA, 256, 0, stream>>>(attn, N, pval, pidx);
    fps_final<<<1, 512, 0, stream>>>(pval, pidx, BA, cur, out, 0, mind, 0);

    // Rounds 1..511: distance sweep + argmax.
    for (int t = 1; t < KSEL; ++t) {
        fps_part<<<BA, 256, 0, stream>>>(feats, mind, cur, N,
                                         (t == 1) ? 1 : 0, pval, pidx);
        fps_final<<<1, 512, 0, stream>>>(pval, pidx, BA, cur, out, t, mind, 1);
    }
}